// BasicGCN_72765335929134
// MI455X (gfx1250) — compile-verified
//
#include <hip/hip_runtime.h>

typedef __attribute__((ext_vector_type(2))) float v2f;
typedef __attribute__((ext_vector_type(8))) float v8f;

#define DK 128        // inner dim for all GEMMs in this net
#define SCAN_B 1024   // elements per scan block

// ---------------------------------------------------------------------------
// Degree count (int): deg[dst[e]] += 1
// ---------------------------------------------------------------------------
__global__ void count_kernel(const int* __restrict__ dst, int* __restrict__ deg, int E) {
    int i = blockIdx.x * blockDim.x + threadIdx.x;
    if (i < E) atomicAdd(&deg[dst[i]], 1);
}

// dis[i] = rsqrt(deg[i] + 1)
__global__ void dis_kernel(const int* __restrict__ deg, float* __restrict__ dis, int n) {
    int i = blockIdx.x * blockDim.x + threadIdx.x;
    if (i < n) dis[i] = rsqrtf((float)deg[i] + 1.0f);
}

// ---------------------------------------------------------------------------
// Exclusive scan of deg[0..n) -> rowptr[0..n); block sums to bsums.
// ---------------------------------------------------------------------------
__global__ __launch_bounds__(SCAN_B)
void scan_block_kernel(const int* __restrict__ in, int* __restrict__ out,
                       int* __restrict__ bsums, int n) {
    __shared__ int tmp[SCAN_B];
    const int t = threadIdx.x;
    const int gid = blockIdx.x * SCAN_B + t;
    int v = (gid < n) ? in[gid] : 0;
    tmp[t] = v;
    __syncthreads();
    for (int off = 1; off < SCAN_B; off <<= 1) {   // inclusive Hillis-Steele
        int add = (t >= off) ? tmp[t - off] : 0;
        __syncthreads();
        tmp[t] += add;
        __syncthreads();
    }
    if (gid < n) out[gid] = tmp[t] - v;            // exclusive
    if (t == SCAN_B - 1) bsums[blockIdx.x] = tmp[t];
}

// Exclusive scan of the (<=1024) block sums, in place.
__global__ __launch_bounds__(SCAN_B)
void scan_sums_kernel(int* __restrict__ bsums, int nb) {
    __shared__ int tmp[SCAN_B];
    const int t = threadIdx.x;
    int v = (t < nb) ? bsums[t] : 0;
    tmp[t] = v;
    __syncthreads();
    for (int off = 1; off < SCAN_B; off <<= 1) {
        int add = (t >= off) ? tmp[t - off] : 0;
        __syncthreads();
        tmp[t] += add;
        __syncthreads();
    }
    if (t < nb) bsums[t] = tmp[t] - v;
}

// rowptr[i] += blockOffset;  rowptr[n] = E
__global__ void scan_add_kernel(int* __restrict__ rowptr, const int* __restrict__ bsums,
                                int n, int total) {
    int gid = blockIdx.x * blockDim.x + threadIdx.x;
    if (gid < n) rowptr[gid] += bsums[gid / SCAN_B];
    if (gid == n) rowptr[n] = total;
}

// CSR fill: csr_src[rowptr[dst[e]] + cursor[dst[e]]++] = src[e]
__global__ void csr_fill_kernel(const int* __restrict__ src, const int* __restrict__ dst,
                                const int* __restrict__ rowptr, int* __restrict__ cursor,
                                int* __restrict__ csr_src, int E) {
    int e = blockIdx.x * blockDim.x + threadIdx.x;
    if (e >= E) return;
    int d = dst[e];
    int pos = atomicAdd(&cursor[d], 1);
    csr_src[rowptr[d] + pos] = src[e];
}

// ---------------------------------------------------------------------------
// C[nrows x DO] = A[nrows x 128] * W[128 x DO] (+ bias), fp32 WMMA 16x16x4.
// ---------------------------------------------------------------------------
template <int DO>
__global__ __launch_bounds__(32 * (DO / 16))
void gcn_gemm_wmma(const float* __restrict__ A, const float* __restrict__ W,
                   const float* __restrict__ bias, float* __restrict__ C, int nrows) {
    constexpr int NW  = DO / 16;   // waves per block
    constexpr int PAD = 132;       // padded LDS row stride (floats)
    __shared__ float sA[16 * PAD];

    const int row0 = blockIdx.x * 16;
    const int tid  = threadIdx.x;
    const int lane = tid & 31;
    const int wave = tid >> 5;
    const int half = lane >> 4;    // 0: lanes 0-15, 1: lanes 16-31
    const int ln   = lane & 15;

    for (int li = tid; li < 16 * DK; li += 32 * NW) {
        int r = li >> 7;
        int c = li & 127;
        int row = row0 + r;
        row = row < nrows ? row : nrows - 1;
        sA[r * PAD + c] = A[(size_t)row * DK + c];
    }
    __syncthreads();

    const int col0 = wave * 16;
    const float* sArow = &sA[ln * PAD];
    v8f acc = {};

#pragma unroll
    for (int k = 0; k < DK; k += 4) {
        const int ka = k + 2 * half;
        v2f a = *(const v2f*)&sArow[ka];
        const float* wp = W + (size_t)ka * DO + col0 + ln;
        v2f b;
        b.x = wp[0];
        b.y = wp[DO];
        acc = __builtin_amdgcn_wmma_f32_16x16x4_f32(
            false, a, false, b, (short)0, acc, false, false);
    }

    const float bv = bias ? bias[col0 + ln] : 0.0f;
#pragma unroll
    for (int r = 0; r < 8; ++r) {
        int row = row0 + r + 8 * half;
        if (row < nrows) C[(size_t)row * DO + col0 + ln] = acc[r] + bv;
    }
}

// ---------------------------------------------------------------------------
// Fused GCN aggregation (atomic-free gather via CSR):
//   out[node] = [relu]( sum_{s in N(node)} tmp[s]*dis[s]*dis[node]
//                       + tmp[node]*dis[node]^2 + bias )
// One block per node, one thread per feature; coalesced 512B row gathers.
// ---------------------------------------------------------------------------
template <int D>
__global__ __launch_bounds__(D)
void aggregate_kernel(const float* __restrict__ tmp, const int* __restrict__ rowptr,
                      const int* __restrict__ csr_src, const float* __restrict__ dis,
                      const float* __restrict__ bias, float* __restrict__ out,
                      int do_relu) {
    const int node = blockIdx.x;
    const int c    = threadIdx.x;
    const float wn = dis[node];
    float acc = tmp[(size_t)node * D + c] * (wn * wn);   // self-loop: 1/deg
    const int j0 = rowptr[node];
    const int j1 = rowptr[node + 1];
    for (int j = j0; j < j1; ++j) {
        int s = csr_src[j];                              // wave-uniform broadcast
        acc += tmp[(size_t)s * D + c] * (dis[s] * wn);
    }
    float v = acc + bias[c];
    out[(size_t)node * D + c] = do_relu ? fmaxf(v, 0.0f) : v;
}

// ---------------------------------------------------------------------------
// Global add pool: out[batch[node]][c] += h[node][c]   (D = 64)
// ---------------------------------------------------------------------------
__global__ void pool_kernel(const float* __restrict__ h, const int* __restrict__ batch,
                            float* __restrict__ out, long long total) {
    long long i = (long long)blockIdx.x * blockDim.x + threadIdx.x;
    if (i >= total) return;
    int node = (int)(i >> 6);
    int c    = (int)(i & 63);
    atomicAdd(&out[(size_t)batch[node] * 64 + c], h[i]);
}

// ---------------------------------------------------------------------------
static inline size_t align256(size_t x) { return (x + 255) & ~(size_t)255; }

extern "C" void kernel_launch(void* const* d_in, const int* in_sizes, int n_in,
                              void* d_out, int out_size, void* d_ws, size_t ws_size,
                              hipStream_t stream) {
    const float* x     = (const float*)d_in[0];
    const int*   ei    = (const int*)d_in[1];
    const int*   batch = (const int*)d_in[2];
    const float* W_emb = (const float*)d_in[3];
    const float* b_emb = (const float*)d_in[4];
    const float* W1 = (const float*)d_in[5];
    const float* b1 = (const float*)d_in[6];
    const float* W2 = (const float*)d_in[7];
    const float* b2 = (const float*)d_in[8];
    const float* W3 = (const float*)d_in[9];
    const float* b3 = (const float*)d_in[10];
    float* out = (float*)d_out;

    const int N = in_sizes[0] / 128;
    const int E = in_sizes[1] / 2;
    const int* src = ei;          // edge_index[0]
    const int* dst = ei + E;      // edge_index[1]
    const int nb = (N + SCAN_B - 1) / SCAN_B;

    // ---- workspace layout ----------------------------------------------
    char* base = (char*)d_ws;
    size_t off = 0;
    int*   deg    = (int*)(base + off); off = align256(off + (size_t)N * 4);
    int*   cursor = (int*)(base + off); off = align256(off + (size_t)N * 4);
    int*   rowptr = (int*)(base + off); off = align256(off + ((size_t)N + 1) * 4);
    int*   bsums  = (int*)(base + off); off = align256(off + (size_t)nb * 4);
    int*   csr    = (int*)(base + off); off = align256(off + (size_t)E * 4);
    float* dis    = (float*)(base + off); off = align256(off + (size_t)N * 4);
    float* hA     = (float*)(base + off); off = align256(off + (size_t)N * 128 * 4);
    float* hB     = (float*)(base + off);

    const int rowTiles = (N + 15) / 16;

    // ---- degree + normalization ----------------------------------------
    hipMemsetAsync(deg, 0, (size_t)N * 4, stream);
    count_kernel<<<(E + 255) / 256, 256, 0, stream>>>(dst, deg, E);
    dis_kernel<<<(N + 255) / 256, 256, 0, stream>>>(deg, dis, N);

    // ---- CSR build: exclusive scan + fill ------------------------------
    scan_block_kernel<<<nb, SCAN_B, 0, stream>>>(deg, rowptr, bsums, N);
    scan_sums_kernel<<<1, SCAN_B, 0, stream>>>(bsums, nb);
    scan_add_kernel<<<(N + 1 + 255) / 256, 256, 0, stream>>>(rowptr, bsums, N, E);
    hipMemsetAsync(cursor, 0, (size_t)N * 4, stream);
    csr_fill_kernel<<<(E + 255) / 256, 256, 0, stream>>>(src, dst, rowptr, cursor, csr, E);

    // ---- embedding: hA = x @ W_emb + b_emb -----------------------------
    gcn_gemm_wmma<128><<<rowTiles, 256, 0, stream>>>(x, W_emb, b_emb, hA, N);

    // ---- GCN layers 1 & 2 (D=128, ReLU), fused aggregate ---------------
    const float* Ws[2] = {W1, W2};
    const float* bs[2] = {b1, b2};
    for (int l = 0; l < 2; ++l) {
        gcn_gemm_wmma<128><<<rowTiles, 256, 0, stream>>>(hA, Ws[l], nullptr, hB, N);
        aggregate_kernel<128><<<N, 128, 0, stream>>>(hB, rowptr, csr, dis, bs[l], hA, 1);
    }

    // ---- GCN layer 3 (D_out=64, no ReLU) -------------------------------
    gcn_gemm_wmma<64><<<rowTiles, 128, 0, stream>>>(hA, W3, nullptr, hB, N);
    aggregate_kernel<64><<<N, 64, 0, stream>>>(hB, rowptr, csr, dis, b3, hA, 0);

    // ---- global add pool -----------------------------------------------
    hipMemsetAsync(out, 0, (size_t)out_size * sizeof(float), stream);
    long long tot = (long long)N * 64;
    pool_kernel<<<(int)((tot + 255) / 256), 256, 0, stream>>>(hA, batch, out, tot);
}